// MultiFrmVQBottleNeck_80788334838456
// MI455X (gfx1250) — compile-verified
//
#include <hip/hip_runtime.h>
#include <hip/hip_bf16.h>

// Problem constants (match reference)
#define BB     16
#define FD     256
#define TT     4096
#define LL     128
#define MM     1024
#define DD     128
#define TCC    2048
#define NROWS  (BB * TCC)   // 32768 rows per codebook
#define NROWSX (BB * TT)    // 65536 (b,t) rows

typedef float v2f __attribute__((ext_vector_type(2)));
typedef float v8f __attribute__((ext_vector_type(8)));
typedef int   v4i __attribute__((vector_size(4 * sizeof(int))));
typedef v4i __attribute__((address_space(1)))* as1_v4i;
typedef v4i __attribute__((address_space(3)))* as3_v4i;

#if defined(__has_builtin)
#if __has_builtin(__builtin_amdgcn_global_load_async_to_lds_b128) && \
    __has_builtin(__builtin_amdgcn_s_wait_asynccnt)
#define HAVE_ASYNC_LDS 1
#endif
#endif

// D = A(16x4,f32) * B(4x16,f32) + C(16x16,f32)   -> v_wmma_f32_16x16x4_f32
static __device__ __forceinline__ v8f wmma_f32(v2f a, v2f b, v8f c) {
  return __builtin_amdgcn_wmma_f32_16x16x4_f32(false, a, false, b,
                                               (short)0, c, false, false);
}

// ---------------- k0: ||e_m||^2 for both codebooks ----------------
__global__ __launch_bounds__(256) void k0_e2(const float* __restrict__ cb,
                                             float* __restrict__ e2) {
  int tid = blockIdx.x * 256 + threadIdx.x;   // 0..2047  (= i*M + m)
  const float* row = cb + (long)tid * DD;
  float s = 0.f;
  #pragma unroll 8
  for (int d = 0; d < DD; ++d) { float v = row[d]; s += v * v; }
  e2[tid] = s;
}

// ---------------- k1: x[b,t,l] = sum_f inputs[b,f,t] * w1[l,f] ----------------
// Block: 8 waves -> 128 rows x 128 cols.  w1 staged in LDS in 64-row K chunks,
// stored K-pair-interleaved so each B fragment is one aligned ds_load_b64.
__global__ __launch_bounds__(256) void k1_gemm1(const float* __restrict__ in,
                                                const float* __restrict__ w1,
                                                float* __restrict__ xbuf) {
  __shared__ float lds[64 * LL];              // 32 KB, pair-interleaved [f/2][l][f%2]
  const int tid = threadIdx.x;
  const int wave = tid >> 5, lane = tid & 31;
  const int m = lane & 15, h = lane >> 4;
  const int rowbase = blockIdx.x * 128 + wave * 16;
  const int rowl = rowbase + m;
  const int bidx = rowl >> 12;                // /T
  const int t    = rowl & (TT - 1);
  const long abase = (long)bidx * FD * TT + t;
  const v2f* ldsp = (const v2f*)lds;

  v8f acc[8];
  #pragma unroll
  for (int ct = 0; ct < 8; ++ct) acc[ct] = {};

  for (int kc = 0; kc < 4; ++kc) {            // 4 chunks of 64 f
    for (int e = tid; e < 64 * LL; e += 256) {
      int fl = e >> 7, l = e & 127;
      // pair-interleaved: float index = (fl/2)*256 + l*2 + (fl&1)
      lds[(fl >> 1) * 256 + l * 2 + (fl & 1)] = w1[l * FD + (kc * 64 + fl)];
    }
    __syncthreads();
    #pragma unroll 4
    for (int ks = 0; ks < 16; ++ks) {
      const int k0 = ks * 4 + 2 * h;          // chunk-local K for this lane half
      const int kg = kc * 64 + k0;            // global f
      v2f a;
      a.x = in[abase + (long)kg * TT];
      a.y = in[abase + (long)(kg + 1) * TT];
      #pragma unroll
      for (int ct = 0; ct < 8; ++ct) {
        const int n = ct * 16 + m;
        v2f b = ldsp[(k0 >> 1) * 128 + n];    // one ds_load_b64
        acc[ct] = wmma_f32(a, b, acc[ct]);
      }
    }
    __syncthreads();
  }
  #pragma unroll
  for (int ct = 0; ct < 8; ++ct)
    #pragma unroll
    for (int r = 0; r < 8; ++r) {
      int row = rowbase + r + 8 * h;
      int col = ct * 16 + m;
      xbuf[(long)row * LL + col] = acc[ct][r];
    }
}

// ---------------- k2: S = x_i @ emb_i^T ; fused score + argmax ----------------
// score = 10*S - 5*||e||^2 + g   (monotone transform of (logits+g)/TAU)
// A fragments (x rows) are tile-invariant: hoisted into 32 v2f registers.
__global__ __launch_bounds__(256) void k2_dist_argmax(
    const float* __restrict__ xbuf, const float* __restrict__ cb,
    const float* __restrict__ gum, const float* __restrict__ e2,
    int* __restrict__ idxw, float* __restrict__ oind) {
  __shared__ float lds[16 * 132];             // emb tile [n][k], padded stride
  const int i = blockIdx.y;                   // codebook
  const int tid = threadIdx.x;
  const int wave = tid >> 5, lane = tid & 31;
  const int m = lane & 15, h = lane >> 4;
  const int rowbase = blockIdx.x * 128 + wave * 16;
  const int rowl = rowbase + m;
  const int bb = rowl >> 11;                  // /TC
  const int tc = rowl & (TCC - 1);
  // x row for (b, t=2tc) at channel block i*64; k>=64 -> next frame (+128 floats)
  const long xb = ((long)bb * TT + 2 * tc) * LL + i * 64;
  const float* embi = cb + (long)i * MM * DD;
  const float* gi = gum + (long)i * NROWS * MM;

  // Hoist A fragments (invariant across all 64 column tiles)
  v2f afrag[32];
  #pragma unroll
  for (int ks = 0; ks < 32; ++ks) {
    int k0 = ks * 4 + 2 * h;                  // k0, k0+1 contiguous in xbuf
    afrag[ks] = *(const v2f*)&xbuf[xb + ((k0 & 64) << 1) + (k0 & 63)];
  }

  float bestv[8]; int besti[8];
  #pragma unroll
  for (int r = 0; r < 8; ++r) { bestv[r] = -3.4e38f; besti[r] = 0; }

  for (int mt = 0; mt < 64; ++mt) {           // 64 column tiles of 16 codes
#ifdef HAVE_ASYNC_LDS
    // async DMA: 512 x 16B chunks, ASYNCcnt-tracked
    for (int e = tid; e < 512; e += 256) {
      int rr = e >> 5, cc = (e & 31) * 4;
      __builtin_amdgcn_global_load_async_to_lds_b128(
          (as1_v4i)(embi + (long)(mt * 16 + rr) * DD + cc),
          (as3_v4i)&lds[rr * 132 + cc],
          0, 0);
    }
    __builtin_amdgcn_s_wait_asynccnt(0);
#else
    for (int e = tid; e < 2048; e += 256) {
      int rr = e >> 7, cc = e & 127;
      lds[rr * 132 + cc] = embi[(long)(mt * 16 + rr) * DD + cc];
    }
#endif
    __syncthreads();
    v8f c = {};
    #pragma unroll 8
    for (int ks = 0; ks < 32; ++ks) {         // K = 128
      int k0 = ks * 4 + 2 * h;
      v2f b = *(const v2f*)&lds[m * 132 + k0];   // contiguous pair, ds_load_b64
      c = wmma_f32(afrag[ks], b, c);
    }
    const int mcol = mt * 16 + m;
    const float e2v = e2[i * MM + mcol];
    #pragma unroll
    for (int r = 0; r < 8; ++r) {
      int nrow = rowbase + r + 8 * h;
      float g = __builtin_nontemporal_load(&gi[(long)nrow * MM + mcol]);
      float sc = 10.0f * c[r] - 5.0f * e2v + g;
      if (sc > bestv[r]) { bestv[r] = sc; besti[r] = mcol; }
    }
    __syncthreads();
  }
  // cross-lane argmax over the 16 lanes holding each row (tie -> lower index)
  #pragma unroll
  for (int r = 0; r < 8; ++r) {
    float v = bestv[r]; int id = besti[r];
    #pragma unroll
    for (int mask = 1; mask <= 8; mask <<= 1) {
      float ov = __shfl_xor(v, mask, 32);
      int   oi = __shfl_xor(id, mask, 32);
      if (ov > v || (ov == v && oi < id)) { v = ov; id = oi; }
    }
    if (m == 0) {
      int nrow = rowbase + r + 8 * h;
      idxw[i * NROWS + nrow] = id;
      oind[(long)nrow * 2 + i] = (float)id;   // quant_inds
    }
  }
}

// ---------------- k2b: vq_feat gather (straight-through hard == emb[idx]) ----
__global__ __launch_bounds__(256) void k2b_gather(const float* __restrict__ cb,
                                                  const int* __restrict__ idxw,
                                                  float* __restrict__ vq) {
  const int tid = threadIdx.x, wave = tid >> 5, lane = tid & 31;
  const int p = blockIdx.x * 8 + wave;        // 0 .. NROWS*2-1
  const int i = p & 1;
  const int n = p >> 1;
  const int id = idxw[i * NROWS + n];
  const int bb = n >> 11, tc = n & (TCC - 1);
  const long dst0 = ((long)bb * TT + 2 * tc) * LL + i * 64;
  const float* src = cb + ((long)i * MM + id) * DD;
  #pragma unroll
  for (int cf = 0; cf < 2; ++cf)
    for (int j = lane; j < 64; j += 32)
      vq[dst0 + cf * LL + j] = src[cf * 64 + j];
}

// ---------------- k3: out[b,f,t] = sum_l vq[b,t,l] * w2[f,l] ----------------
// w2 tile pair-interleaved in LDS; A row pairs contiguous -> v2f loads.
__global__ __launch_bounds__(256) void k3_gemm2(const float* __restrict__ vq,
                                                const float* __restrict__ w2,
                                                float* __restrict__ out) {
  __shared__ float lds[LL * 64];              // 32 KB, pair-interleaved [l/2][f][l%2]
  const int tid = threadIdx.x, wave = tid >> 5, lane = tid & 31;
  const int m = lane & 15, h = lane >> 4;
  const int rowbase = blockIdx.x * 128 + wave * 16;
  const int colbase = blockIdx.y * 64;
  const v2f* ldsp = (const v2f*)lds;
  for (int e = tid; e < LL * 64; e += 256) {
    int l = e >> 6, f = e & 63;
    // pair-interleaved: float index = (l/2)*128 + f*2 + (l&1)
    lds[(l >> 1) * 128 + f * 2 + (l & 1)] = w2[(long)(colbase + f) * LL + l];
  }
  __syncthreads();
  const long arow = (long)(rowbase + m) * LL;
  v8f acc[4];
  #pragma unroll
  for (int ct = 0; ct < 4; ++ct) acc[ct] = {};
  #pragma unroll 4
  for (int ks = 0; ks < 32; ++ks) {           // K = 128
    int k0 = ks * 4 + 2 * h;
    v2f a = *(const v2f*)&vq[arow + k0];      // contiguous pair
    #pragma unroll
    for (int ct = 0; ct < 4; ++ct) {
      int n = ct * 16 + m;
      v2f b = ldsp[(k0 >> 1) * 64 + n];       // one ds_load_b64
      acc[ct] = wmma_f32(a, b, acc[ct]);
    }
  }
  #pragma unroll
  for (int ct = 0; ct < 4; ++ct)
    #pragma unroll
    for (int r = 0; r < 8; ++r) {
      int row = rowbase + r + 8 * h;
      int bb = row >> 12, t = row & (TT - 1);
      int f = colbase + ct * 16 + m;
      out[((long)bb * FD + f) * TT + t] = acc[ct][r];
    }
}

extern "C" void kernel_launch(void* const* d_in, const int* in_sizes, int n_in,
                              void* d_out, int out_size, void* d_ws, size_t ws_size,
                              hipStream_t stream) {
  const float* in  = (const float*)d_in[0];   // [B,F,T,1]
  const float* w1  = (const float*)d_in[1];   // [L,F]
  const float* w2  = (const float*)d_in[2];   // [F,L]
  const float* cb  = (const float*)d_in[3];   // [K,M,D]
  const float* gum = (const float*)d_in[4];   // [K,B*TC,M]

  float* out  = (float*)d_out;                        // [B,F,T,1]
  float* vq   = out + (long)BB * FD * TT;             // [B,T,L]
  float* oind = vq + (long)BB * TT * LL;              // [B,TC,K]

  float* xbuf = (float*)d_ws;                         // [B*T, L]
  float* e2   = xbuf + (long)NROWSX * LL;             // [K*M]
  int*   idxw = (int*)(e2 + 2 * MM);                  // [K*NROWS]

  k0_e2<<<(2 * MM) / 256, 256, 0, stream>>>(cb, e2);
  k1_gemm1<<<NROWSX / 128, 256, 0, stream>>>(in, w1, xbuf);
  dim3 g2(NROWS / 128, 2);
  k2_dist_argmax<<<g2, 256, 0, stream>>>(xbuf, cb, gum, e2, idxw, oind);
  k2b_gather<<<(NROWS * 2) / 8, 256, 0, stream>>>(cb, idxw, vq);
  dim3 g3(NROWSX / 128, 4);
  k3_gemm2<<<g3, 256, 0, stream>>>(vq, w2, out);
}